// AttNHP_46952582479930
// MI455X (gfx1250) — compile-verified
//
#include <hip/hip_runtime.h>
#include <math.h>

// ---------------------------------------------------------------------------
// AttNHP forward for gfx1250 (MI455X). fp32 end-to-end using
// V_WMMA_F32_16X16X4_F32 for every matmul; flash-style online-softmax
// attention so the 2048x2048 score matrix is never materialized.
// ---------------------------------------------------------------------------

namespace {

constexpr int   kT   = 1024;   // T
constexpr int   kTT  = 2048;   // 2T
constexpr int   kDK  = 256;    // d_type
constexpr int   kB   = 4;      // batch
constexpr float kNeg     = -1000000000.0f;  // mask fill (finite, like reference)
constexpr float kScale   = 0.0625f;         // 1/sqrt(256)
constexpr float kMInit   = -3.0e38f;        // running-max init (finite, avoids inf-inf)

typedef __attribute__((ext_vector_type(2))) float v2f;
typedef __attribute__((ext_vector_type(8))) float v8f;

__device__ __forceinline__ v8f wmma_f32(v2f a, v2f b, v8f c) {
  // D = A(16x4,f32) * B(4x16,f32) + C(16x16,f32)
  return __builtin_amdgcn_wmma_f32_16x16x4_f32(false, a, false, b, (short)0, c,
                                               false, false);
}

__device__ __forceinline__ v8f v8zero() {
  v8f r;
#pragma unroll
  for (int i = 0; i < 8; ++i) r[i] = 0.0f;
  return r;
}

// ---------------- kernel 1: type/time embeddings, cur = 0 ------------------
__global__ __launch_bounds__(256) void init_kernel(
    const float* __restrict__ event_seqs,  // [B,T,64]
    const float* __restrict__ time_seqs,   // [B,T]
    const float* __restrict__ Wtype,       // [64,256]
    const float* __restrict__ btype,       // [256]
    float* __restrict__ evt,               // [B,T,256] (type_emb, later out[:, :T])
    float* __restrict__ temb,              // [B,T,256]
    float* __restrict__ cur) {             // [B,T,256]
  const int t = blockIdx.x, b = blockIdx.y, d = threadIdx.x;
  const size_t row = (size_t)b * kT + t;
  const float* e = event_seqs + row * 64;
  float s = btype[d];
#pragma unroll 8
  for (int j = 0; j < 64; ++j) s += e[j] * Wtype[j * kDK + d];
  const size_t idx = row * kDK + d;
  evt[idx] = tanhf(s);
  cur[idx] = 0.0f;
  if (d < 128) {
    const float tv  = time_seqs[row];
    const float div = __expf((float)(2 * d) * (-logf(10000.0f) / 256.0f));
    const float ang = tv * div;
    temb[row * kDK + 2 * d]     = __sinf(ang);
    temb[row * kDK + 2 * d + 1] = __cosf(ang);
  }
}

__global__ __launch_bounds__(256) void zero_kernel(float* __restrict__ p) {
  p[(size_t)blockIdx.x * 256 + threadIdx.x] = 0.0f;
}

// ---------------- kernel 2: Q/K/V projections via f32 WMMA -----------------
// x[b,r,:] = r<T ? concat(evt[b,r], temb[b,r]) : concat(cur[b,r-T], temb[b,r-T])
// One wave computes a 16(M) x 64(N) output strip; K=512 in 128 WMMA steps.
__global__ __launch_bounds__(32) void qkv_kernel(
    const float* __restrict__ evt, const float* __restrict__ cur,
    const float* __restrict__ temb,
    const float* __restrict__ Wq, const float* __restrict__ Wk,
    const float* __restrict__ Wv,
    const float* __restrict__ bq, const float* __restrict__ bk,
    const float* __restrict__ bv,
    float* __restrict__ q, float* __restrict__ k, float* __restrict__ v,
    int hl) {
  const int lane = threadIdx.x;
  const int nl = lane & 15;          // A row / B col within tile
  const int kb = (lane >> 4) * 2;    // K sub-offset for this half-wave

  const float* W; const float* bias; float* outp;
  if (blockIdx.z == 0)      { W = Wq; bias = bq; outp = q; }
  else if (blockIdx.z == 1) { W = Wk; bias = bk; outp = k; }
  else                      { W = Wv; bias = bv; outp = v; }
  W    += (size_t)hl * 512 * kDK;    // Wx[h,l]  (512 x 256)
  bias += (size_t)hl * kDK;

  const int gr0 = blockIdx.x * 16;   // global row in [0, B*2T)
  const int b   = gr0 / kTT;
  const int r0  = gr0 % kTT;
  const bool top = (r0 < kT);
  const int  rr0 = top ? r0 : (r0 - kT);
  const float* left = (top ? evt : cur) + ((size_t)b * kT + rr0 + nl) * kDK;
  const float* tim  = temb              + ((size_t)b * kT + rr0 + nl) * kDK;
  const int nbase = blockIdx.y * 64;

  v8f acc[4];
#pragma unroll
  for (int t = 0; t < 4; ++t) acc[t] = v8zero();

  for (int s = 0; s < 128; ++s) {
    const int kk = s * 4 + kb;       // uniform branch: kk<256 for s<64
    const v2f a = *(const v2f*)(kk < 256 ? (left + kk) : (tim + (kk - 256)));
#pragma unroll
    for (int t = 0; t < 4; ++t) {
      const int n = nbase + t * 16 + nl;
      v2f bb;
      bb.x = W[(size_t)kk * kDK + n];
      bb.y = W[(size_t)(kk + 1) * kDK + n];
      acc[t] = wmma_f32(a, bb, acc[t]);
    }
  }
  const int rowoff = (lane >> 4) * 8;
#pragma unroll
  for (int t = 0; t < 4; ++t) {
    const int n = nbase + t * 16 + nl;
    const float bi = bias[n];
#pragma unroll
    for (int i = 0; i < 8; ++i) {
      const int r = r0 + rowoff + i;  // C/D layout: vgpr i -> M = i + 8*hi
      outp[((size_t)b * kTT + r) * kDK + n] = acc[t][i] + bi;
    }
  }
}

// ---------------- kernel 3: flash attention (masked softmax) ---------------
// One wave owns a 16-row query tile and streams 128 key tiles of 16.
__global__ __launch_bounds__(32) void attn_kernel(
    const float* __restrict__ q, const float* __restrict__ k,
    const float* __restrict__ v, const unsigned char* __restrict__ npm,
    float* __restrict__ out) {
  __shared__ float ldsQ[16 * 260];   // padded stride: conflict-free A reads
  __shared__ float ldsP[16 * 17];    // P tile transit (C/D -> A layout)
  const int lane = threadIdx.x;
  const int nl = lane & 15;
  const int hi = lane >> 4;
  const int kb = hi * 2;
  const int b  = blockIdx.y;
  const int r0 = blockIdx.x * 16;
  const bool top = (r0 < kT);
  const int rbase = r0 + 8 * hi;     // rows carried by this lane's acc regs

  // stage Q tile (16 x 256) in LDS once; reused for all 128 key tiles
  for (int i = lane; i < 16 * 64; i += 32) {
    const int row = i >> 6;
    const int c4  = (i & 63) * 4;
    const float4 val =
        *(const float4*)(q + ((size_t)b * kTT + r0 + row) * kDK + c4);
    *(float4*)(&ldsQ[row * 260 + c4]) = val;
  }
  __syncthreads();

  float mrun[8], lrun[8];
#pragma unroll
  for (int i = 0; i < 8; ++i) { mrun[i] = kMInit; lrun[i] = 0.0f; }
  v8f acc[16];                       // 16 x 256 f32 output accumulator
#pragma unroll
  for (int t = 0; t < 16; ++t) acc[t] = v8zero();

  for (int kt = 0; kt < 128; ++kt) {
    const int c0 = kt * 16;
    const int c  = c0 + nl;          // key/column index for this lane

    {  // prefetch next K/V tiles (global_prefetch_b8)
      const int cn = ((kt + 1) < 128 ? (kt + 1) : kt) * 16 + nl;
      __builtin_prefetch(k + ((size_t)b * kTT + cn) * kDK, 0, 1);
      __builtin_prefetch(v + ((size_t)b * kTT + cn) * kDK, 0, 1);
    }

    // ---- S = Q * K^T : A from LDS, B straight from global (read-once) ----
    const float* krow = k + ((size_t)b * kTT + c) * kDK;
    v8f s = v8zero();
    for (int st = 0; st < 64; ++st) {
      const int d = st * 4 + kb;
      const v2f a  = *(const v2f*)(&ldsQ[nl * 260 + d]);
      const v2f bb = *(const v2f*)(krow + d);
      s = wmma_f32(a, bb, s);
    }

    // ---- scale + mask + online softmax (row stats via 16-lane shuffles) --
    const bool cLeft = (c0 < kT);
    float pv[8], alpha[8];
#pragma unroll
    for (int i = 0; i < 8; ++i) {
      const int r = rbase + i;
      float val = s[i] * kScale;
      bool masked;
      if (cLeft) {                                // cols [0,T)
        const int rr = top ? r : (r - kT);
        masked = (c >= rr) && (npm[((size_t)b * kT + rr) * kT + c] != 0);
      } else {                                    // cols [T,2T)
        masked = top ? true : (c != r);           // top: all; bottom: ~eye
      }
      val = masked ? kNeg : val;
      float mx = val;
      mx = fmaxf(mx, __shfl_xor(mx, 1, 32));
      mx = fmaxf(mx, __shfl_xor(mx, 2, 32));
      mx = fmaxf(mx, __shfl_xor(mx, 4, 32));
      mx = fmaxf(mx, __shfl_xor(mx, 8, 32));
      const float mn = fmaxf(mrun[i], mx);
      const float p  = __expf(val - mn);
      float rs = p;
      rs += __shfl_xor(rs, 1, 32);
      rs += __shfl_xor(rs, 2, 32);
      rs += __shfl_xor(rs, 4, 32);
      rs += __shfl_xor(rs, 8, 32);
      alpha[i] = __expf(mrun[i] - mn);
      lrun[i]  = lrun[i] * alpha[i] + rs;
      mrun[i]  = mn;
      pv[i]    = p;
    }
#pragma unroll
    for (int t = 0; t < 16; ++t)
#pragma unroll
      for (int i = 0; i < 8; ++i) acc[t][i] *= alpha[i];

    // ---- stage P through LDS: C/D layout -> A layout ----
    __syncthreads();
#pragma unroll
    for (int i = 0; i < 8; ++i) ldsP[(i + 8 * hi) * 17 + nl] = pv[i];
    __syncthreads();

    // ---- O += P * V : V consumed from global in B layout (read-once) ----
    const float* vbase = v + ((size_t)b * kTT + c0) * kDK;
#pragma unroll
    for (int ks = 0; ks < 4; ++ks) {
      const int kk = ks * 4 + kb;
      v2f a;
      a.x = ldsP[nl * 17 + kk];
      a.y = ldsP[nl * 17 + kk + 1];
      const float* v0 = vbase + (size_t)kk * kDK + nl;
#pragma unroll
      for (int t = 0; t < 16; ++t) {
        v2f bb;
        bb.x = v0[t * 16];
        bb.y = v0[t * 16 + kDK];
        acc[t] = wmma_f32(a, bb, acc[t]);
      }
    }
    __syncthreads();
  }

  // ---- normalize by softmax denominator and write out ----
#pragma unroll
  for (int t = 0; t < 16; ++t) {
    const int d = t * 16 + nl;
#pragma unroll
    for (int i = 0; i < 8; ++i) {
      const int r = rbase + i;
      out[((size_t)b * kTT + r) * kDK + d] = acc[t][i] / lrun[i];
    }
  }
}

// ---------------- kernel 4: residual tanh + LayerNorm + outputs ------------
__global__ __launch_bounds__(256) void update_kernel(
    const float* __restrict__ attn_out,  // [B,2T,256]
    float* __restrict__ evt, float* __restrict__ cur,
    const float* __restrict__ nw, const float* __restrict__ nb,
    float* __restrict__ dout, int write_out, int h) {
  __shared__ float red[256];
  const int r = blockIdx.x, b = blockIdx.y, d = threadIdx.x;
  const size_t row = (size_t)b * kT + r;
  const size_t idx = row * kDK + d;
  evt[idx] = attn_out[((size_t)b * kTT + r) * kDK + d];          // out[:, :T]
  const float o  = attn_out[((size_t)b * kTT + kT + r) * kDK + d];
  const float cn = tanhf(o) + cur[idx];
  red[d] = cn;
  __syncthreads();
  for (int off = 128; off > 0; off >>= 1) {
    if (d < off) red[d] += red[d + off];
    __syncthreads();
  }
  const float mu = red[0] * (1.0f / 256.0f);
  __syncthreads();
  const float xm = cn - mu;
  red[d] = xm * xm;
  __syncthreads();
  for (int off = 128; off > 0; off >>= 1) {
    if (d < off) red[d] += red[d + off];
    __syncthreads();
  }
  const float var = red[0] * (1.0f / 256.0f);
  const float y = xm * rsqrtf(var + 1e-5f) * nw[d] + nb[d];
  cur[idx] = y;
  if (write_out) dout[row * (kDK * 4) + h * kDK + d] = y;        // concat heads
}

}  // namespace

// ---------------------------------------------------------------------------
extern "C" void kernel_launch(void* const* d_in, const int* in_sizes, int n_in,
                              void* d_out, int out_size, void* d_ws,
                              size_t ws_size, hipStream_t stream) {
  (void)in_sizes; (void)n_in; (void)out_size; (void)ws_size;
  const float* event_seqs = (const float*)d_in[0];
  const float* time_seqs  = (const float*)d_in[1];
  const unsigned char* npm = (const unsigned char*)d_in[2];  // bool bytes
  const float* Wtype = (const float*)d_in[3];
  const float* btype = (const float*)d_in[4];
  const float* Wq = (const float*)d_in[5];
  const float* bq = (const float*)d_in[6];
  const float* Wk = (const float*)d_in[7];
  const float* bk = (const float*)d_in[8];
  const float* Wv = (const float*)d_in[9];
  const float* bv = (const float*)d_in[10];
  const float* nw = (const float*)d_in[11];
  const float* nb = (const float*)d_in[12];
  float* dout = (float*)d_out;

  // workspace layout (floats): all iteration state lives in L2 (192 MB)
  float* ws   = (float*)d_ws;
  const size_t nBT  = (size_t)kB * kT * kDK;    // 1,048,576
  const size_t nBTT = (size_t)kB * kTT * kDK;   // 2,097,152
  float* temb = ws;
  float* evt  = temb + nBT;
  float* cur  = evt + nBT;
  float* qb   = cur + nBT;
  float* kb   = qb + nBTT;
  float* vb   = kb + nBTT;
  float* ob   = vb + nBTT;

  const dim3 gRow(kT, kB);   // (row, batch), 256 threads
  init_kernel<<<gRow, 256, 0, stream>>>(event_seqs, time_seqs, Wtype, btype,
                                        evt, temb, cur);
  for (int h = 0; h < 4; ++h) {
    if (h > 0) zero_kernel<<<dim3((unsigned)(nBT / 256)), 256, 0, stream>>>(cur);
    for (int l = 0; l < 3; ++l) {
      const int hl = h * 3 + l;
      qkv_kernel<<<dim3(kB * kTT / 16, 4, 3), 32, 0, stream>>>(
          evt, cur, temb, Wq, Wk, Wv, bq, bk, bv, qb, kb, vb, hl);
      attn_kernel<<<dim3(kTT / 16, kB), 32, 0, stream>>>(qb, kb, vb, npm, ob);
      update_kernel<<<gRow, 256, 0, stream>>>(ob, evt, cur, nw, nb, dout,
                                              (l == 2) ? 1 : 0, h);
    }
  }
}